// RayTransformer_77300821394056
// MI455X (gfx1250) — compile-verified
//
#include <hip/hip_runtime.h>

typedef __attribute__((ext_vector_type(2))) float v2f;
typedef __attribute__((ext_vector_type(8))) float v8f;

#define B_ 32
#define R_ 3136
#define O_ 20
#define D_ 32
#define RAYS_PER_BLOCK 64
#define THREADS 256

// One block: 64 rays of one batch. 8 waves = 4 ray-groups x 2 d-halves.
// Phase 1: stage depth/trans -> LDS (coalesced).
// Phase 2: ponderation (VALU v_exp_f32), write to LDS + global (coalesced).
// Phase 3: per-ray 1x20 . 20x32 matvec via V_WMMA_F32_16X16X4_F32
//          (ponderation replicated across A rows; 5 K-chunks of 4).
// Phase 4: LDS-transposed coalesced stores of result [B,D,H,W].
__global__ __launch_bounds__(THREADS)
void ray_composite_kernel(const float* __restrict__ F,      // [B,R,O,D]
                          const float* __restrict__ depth,  // [B,R,O]
                          const float* __restrict__ trans,  // [B,R,O]
                          float* __restrict__ outRes,       // [B,D,R]
                          float* __restrict__ outPond)      // [B,R,O]
{
    __shared__ float lds_depth[RAYS_PER_BLOCK * O_];
    __shared__ float lds_trans[RAYS_PER_BLOCK * O_];
    __shared__ float lds_pond [RAYS_PER_BLOCK * O_];
    __shared__ float lds_res  [RAYS_PER_BLOCK * 33];   // 33-pad kills bank conflicts

    const int t  = threadIdx.x;
    const int b  = blockIdx.y;
    const int r0 = blockIdx.x * RAYS_PER_BLOCK;
    const size_t rayBase = (size_t)b * R_ + r0;

    // ---- Phase 1: stage depth / transmittance (1280 floats each, contiguous)
    const float* gD = depth + rayBase * O_;
    const float* gT = trans + rayBase * O_;
    #pragma unroll
    for (int i = 0; i < (RAYS_PER_BLOCK * O_) / THREADS; ++i) {
        int idx = i * THREADS + t;
        lds_depth[idx] = gD[idx];
        lds_trans[idx] = gT[idx];
    }
    __syncthreads();

    // ---- Phase 2: ponderation
    float* gP = outPond + rayBase * O_;
    #pragma unroll
    for (int i = 0; i < (RAYS_PER_BLOCK * O_) / THREADS; ++i) {
        int idx = i * THREADS + t;          // flat (ray, object)
        int rr  = idx / O_;
        int o   = idx - rr * O_;
        float d_o = lds_depth[rr * O_ + o];
        float t_o = lds_trans[rr * O_ + o];
        float acc = 0.0f;
        #pragma unroll
        for (int o2 = 0; o2 < O_; ++o2) {
            float diff = d_o - lds_depth[rr * O_ + o2];
            // sigmoid(1000*diff); saturates cleanly via exp->inf / exp->0
            float s = 1.0f / (1.0f + __expf(-1000.0f * diff));
            s = (o2 == o) ? 0.0f : s;       // zero diagonal
            acc += lds_trans[rr * O_ + o2] * s;
        }
        float p = __expf(acc) * (1.0f - __expf(t_o));
        lds_pond[idx] = p;
        gP[idx] = p;                        // coalesced 1KB bursts
    }
    __syncthreads();

    // ---- Phase 3: WMMA f32 16x16x4 per-ray matvec
    const int wave   = t >> 5;              // 0..7
    const int lane   = t & 31;
    const int grp    = wave >> 1;           // ray group 0..3
    const int dbase  = (wave & 1) * 16;     // d-half
    const int rrBase = grp * 16;
    const int n      = lane & 15;           // N index (d within half)
    const int khalf  = (lane >> 4) << 1;    // lanes 0-15 -> K 0,1 ; 16-31 -> K 2,3

    for (int i = 0; i < 16; ++i) {
        const int rr = rrBase + i;
        const float* Fr = F + (rayBase + rr) * (size_t)(O_ * D_);
        if (i < 15) {
            // pull next ray's tile toward the caches (global_prefetch_b8)
            __builtin_prefetch(Fr + O_ * D_ + khalf * D_ + dbase + n, 0, 1);
        }
        v8f c = {};
        #pragma unroll
        for (int kc = 0; kc < 5; ++kc) {    // K = 20 = 5 chunks of 4
            int k0 = kc * 4 + khalf;
            v2f a  = { lds_pond[rr * O_ + k0],
                       lds_pond[rr * O_ + k0 + 1] };        // replicated A rows
            v2f bf = { Fr[(size_t)k0 * D_ + dbase + n],     // 64B coalesced rows
                       Fr[(size_t)(k0 + 1) * D_ + dbase + n] };
            c = __builtin_amdgcn_wmma_f32_16x16x4_f32(
                    false, a, false, bf, (short)0, c, false, false);
        }
        // all C rows identical; c[0] at lane n holds result[dbase+n]
        if (lane < 16)
            lds_res[rr * 33 + dbase + n] = c[0];
    }
    __syncthreads();

    // ---- Phase 4: coalesced transposed stores: out[b, d, r0+rr]
    #pragma unroll
    for (int ch = 0; ch < (RAYS_PER_BLOCK * D_) / THREADS; ++ch) {
        int flat = ch * THREADS + t;
        int d    = flat >> 6;               // /64
        int rr   = flat & 63;               // consecutive lanes -> contiguous 256B
        outRes[((size_t)b * D_ + d) * R_ + r0 + rr] = lds_res[rr * 33 + d];
    }
}

extern "C" void kernel_launch(void* const* d_in, const int* in_sizes, int n_in,
                              void* d_out, int out_size, void* d_ws, size_t ws_size,
                              hipStream_t stream) {
    (void)in_sizes; (void)n_in; (void)out_size; (void)d_ws; (void)ws_size;
    const float* F     = (const float*)d_in[0];   // form_decoding [B,R,O,D]
    const float* depth = (const float*)d_in[1];   // [B,R,O]
    const float* trans = (const float*)d_in[2];   // [B,R,O]
    float* outRes  = (float*)d_out;                         // [B,D,H,W] flat
    float* outPond = outRes + (size_t)B_ * D_ * R_;         // [B,R,O] flat
    dim3 grid(R_ / RAYS_PER_BLOCK, B_);
    ray_composite_kernel<<<grid, THREADS, 0, stream>>>(F, depth, trans, outRes, outPond);
}